// Iterative_9174050144276
// MI455X (gfx1250) — compile-verified
//
#include <hip/hip_runtime.h>
#include <math.h>

// ---------------- problem constants (match reference) ----------------
#define Hc   480
#define Wc   640
#define HWc  (Hc * Wc)
#define Bc   4
#define Pc   4
#define Nc   262144
#define BHWc (Bc * HWc)

#define TILE_H 16   // rows processed per spatial block (plus 1 halo row in LDS)

// accumulator slot offsets (floats) inside the accum region of d_ws
#define ACC_FOCUS_SQ  0                 // [Bc]
#define ACC_FOCUS_CNT 4                 // [Bc]
#define ACC_SPAT      8                 // [Bc*Pc*4]  (dx,dy,dr,ur per (b,p))
#define ACC_TEMP      72                // [Bc*(Pc-1)*2] (dt_sum, mask_sum)
#define ACC_SIZE      128

typedef __attribute__((ext_vector_type(2))) float        v2f;
typedef __attribute__((ext_vector_type(8))) float        v8f;
typedef __attribute__((ext_vector_type(4))) unsigned int v4u;
typedef __attribute__((ext_vector_type(8))) int          v8i;
typedef __attribute__((ext_vector_type(4))) int          v4i;

// -------- wave32 sum reduction via V_WMMA_F32_16X16X4_F32 ------------
// A(16x4): lanes 0-15 hold K=0,1 (vgpr0,1); lanes 16-31 hold K=2,3.
// Put lane partial in vgpr0, 0 in vgpr1; B = ones => D[m][*] = x_m + x_{m+16}.
// C/D: lanes 0-15 carry rows 0-7 in c[0..7], lanes 16-31 rows 8-15.
__device__ __forceinline__ float wave_sum32(float x) {
#if __has_builtin(__builtin_amdgcn_wmma_f32_16x16x4_f32)
  v2f a; a[0] = x;    a[1] = 0.0f;
  v2f b; b[0] = 1.0f; b[1] = 1.0f;
  v8f c = {};
  c = __builtin_amdgcn_wmma_f32_16x16x4_f32(
      /*neg_a=*/false, a, /*neg_b=*/false, b,
      /*c_mod=*/(short)0, c, /*reuse_a=*/false, /*reuse_b=*/false);
  float s = c[0] + c[1] + c[2] + c[3] + c[4] + c[5] + c[6] + c[7];
  s += __shfl_xor(s, 16, 32);   // rows 0-7 half + rows 8-15 half
  return s;
#else
  for (int off = 16; off > 0; off >>= 1) x += __shfl_xor(x, off, 32);
  return x;
#endif
}

__device__ __forceinline__ float charb(float a) { return sqrtf(a * a + 1e-6f); }

// -------- Tensor Data Mover: DMA a rows x 640 f32 tile into LDS ------
// D# packing per CDNA5 ISA ch.8 (group0: count/lds_addr/global_addr/type,
// group1: data_size, tensor_dim0/1, tile_dim0/1, tensor_dim0_stride).
#if __has_builtin(__builtin_amdgcn_tensor_load_to_lds)
#define HAVE_TDM 1
__device__ __forceinline__ void tdm_load_rows(const float* gsrc, unsigned lds_off,
                                              int tensor_rows, int tile_rows) {
  unsigned long long ga = (unsigned long long)(uintptr_t)gsrc;
  v4u g0;
  g0[0] = 1u;                                            // count=1 (valid user D#)
  g0[1] = lds_off;                                       // lds_addr (bytes)
  g0[2] = (unsigned)(ga & 0xFFFFFFFFu);                  // global_addr[31:0]
  g0[3] = (unsigned)((ga >> 32) & 0x01FFFFFFu)           // global_addr[56:32]
          | (2u << 30);                                  // type=2 ("image")
  v8i g1;
  g1[0] = (int)(2u << 16);                               // data_size=2 (4 bytes)
  g1[1] = (int)((unsigned)Wc << 16);                     // tensor_dim0[15:0]=640
  g1[2] = (int)(((unsigned)tensor_rows & 0xFFFFu) << 16);// td0[31:16]=0 | td1[15:0]
  g1[3] = (int)((unsigned)Wc << 16);                     // td1[31:16]=0 | tile_dim0=640
  g1[4] = (int)((unsigned)tile_rows & 0xFFFFu);          // tile_dim1 | tile_dim2=0
  g1[5] = Wc;                                            // tensor_dim0_stride[31:0]
  g1[6] = 0;                                             // stride0[47:32] | td1_stride lo
  g1[7] = 0;
  v4i gz = {0, 0, 0, 0};
#if __clang_major__ >= 23
  v8i gz8 = {0, 0, 0, 0, 0, 0, 0, 0};
  __builtin_amdgcn_tensor_load_to_lds(g0, g1, gz, gz, gz8, 0);
#else
  __builtin_amdgcn_tensor_load_to_lds(g0, g1, gz, gz, 0);
#endif
}
#else
#define HAVE_TDM 0
#endif

// ---------------- kernels -------------------------------------------
__global__ void k_zero(float* __restrict__ p, int n) {
  int i = blockIdx.x * blockDim.x + threadIdx.x;
  int stride = gridDim.x * blockDim.x;
  for (; i < n; i += stride) p[i] = 0.0f;
}

// Bilinear scatter of events into 4 planes: [pos, neg, pos_ts, neg_ts], each Bc*HWc.
__global__ void k_scatter(const float* __restrict__ ev,
                          const float* __restrict__ pol_mask,
                          const float* __restrict__ ts_list,
                          float* __restrict__ iwe) {
  int gid = blockIdx.x * blockDim.x + threadIdx.x;
  if (gid >= Bc * Nc) return;
  int b = gid / Nc;
  int n = gid - b * Nc;

  // stream prefetch of upcoming event coordinates (-> global_prefetch_b8)
  __builtin_prefetch(ev + (size_t)gid * 2 + 4096, 0, 1);

  float y = ev[(size_t)gid * 2 + 0];
  float x = ev[(size_t)gid * 2 + 1];
  float pol = pol_mask[((size_t)b * (4 * Nc) + n) * 2 + 0];  // tiled: first N rows
  float ts  = ts_list[(size_t)b * (4 * Nc) + n];
  float norm_ts = 1.0f - fabsf(1.0f - ts);  // TREF=1, TS_SCALING=1
  float ts2 = norm_ts * norm_ts;            // both weight components scaled by norm_ts

  float y0f = floorf(y), x0f = floorf(x);
  int iy0 = (int)y0f, ix0 = (int)x0f;
  float fy = y - y0f, fx = x - x0f;

  float* plane    = iwe + ((pol > 0.5f) ? 0 : 1) * (size_t)BHWc + (size_t)b * HWc;
  float* plane_ts = plane + 2 * (size_t)BHWc;

  float wys[2] = {1.0f - fy, fy}; int cys[2] = {iy0, iy0 + 1};
  float wxs[2] = {1.0f - fx, fx}; int cxs[2] = {ix0, ix0 + 1};
#pragma unroll
  for (int cy = 0; cy < 2; ++cy)
#pragma unroll
    for (int cx = 0; cx < 2; ++cx) {
      int Y = cys[cy], X = cxs[cx];
      if (Y >= 0 && Y < Hc && X >= 0 && X < Wc) {
        float wp = wys[cy] * wxs[cx];
        int off = Y * Wc + X;
        atomicAdd(plane + off, wp);
        atomicAdd(plane_ts + off, wp * ts2);
      }
    }
}

// Focus loss partials: per batch, sum(pos_ts^2 + neg_ts^2) and nonzero-pixel count.
__global__ void k_focus(const float* __restrict__ iwe, float* __restrict__ acc) {
  const int b = blockIdx.y;
  const float* pos = iwe + (size_t)b * HWc;
  const float* neg = pos + (size_t)BHWc;
  const float* pts = pos + 2 * (size_t)BHWc;
  const float* nts = pos + 3 * (size_t)BHWc;
  float sq = 0.0f, cnt = 0.0f;
  for (int i = blockIdx.x * blockDim.x + threadIdx.x; i < HWc;
       i += gridDim.x * blockDim.x) {
    float a = pts[i], c = nts[i];
    sq += a * a + c * c;
    cnt += ((pos[i] + neg[i]) > 0.0f) ? 1.0f : 0.0f;
  }
  sq = wave_sum32(sq);
  cnt = wave_sum32(cnt);
  if ((threadIdx.x & 31) == 0) {
    atomicAdd(&acc[ACC_FOCUS_SQ + b], sq);
    atomicAdd(&acc[ACC_FOCUS_CNT + b], cnt);
  }
}

// Spatial Charbonnier smoothing, 4 terms per (b,p) plane.
// TDM streams a (TILE_H+1)x640 tile of fx and fy into LDS; stencil reads hit LDS.
__global__ void k_spatial(const float* __restrict__ flow, float* __restrict__ acc) {
  const int bp = blockIdx.y;  // b*Pc + p
  const int r0 = blockIdx.x * TILE_H;
  const float* fx = flow + ((size_t)bp * 2 + 0) * HWc;
  const float* fy = flow + ((size_t)bp * 2 + 1) * HWc;

  __shared__ float lfx[(TILE_H + 1) * Wc];
  __shared__ float lfy[(TILE_H + 1) * Wc];
  const int rows_avail = Hc - r0;
  const int rows_tile  = (rows_avail < TILE_H + 1) ? rows_avail : (TILE_H + 1);

#if HAVE_TDM
  if (threadIdx.x < 32) {  // wave 0 issues both DMAs (TDM ignores EXEC)
    tdm_load_rows(fx + (size_t)r0 * Wc, (unsigned)(uintptr_t)&lfx[0],
                  rows_avail, rows_tile);
    tdm_load_rows(fy + (size_t)r0 * Wc, (unsigned)(uintptr_t)&lfy[0],
                  rows_avail, rows_tile);
    __builtin_amdgcn_s_wait_tensorcnt(0);
  }
  __syncthreads();
#else
  for (int t = threadIdx.x; t < rows_tile * Wc; t += blockDim.x) {
    lfx[t] = fx[(size_t)r0 * Wc + t];
    lfy[t] = fy[(size_t)r0 * Wc + t];
  }
  __syncthreads();
#endif

  float sdx = 0.0f, sdy = 0.0f, sdr = 0.0f, sur = 0.0f;
  for (int t = threadIdx.x; t < TILE_H * Wc; t += blockDim.x) {
    int ly = t / Wc, x = t - ly * Wc;
    int y  = r0 + ly;
    float fx00 = lfx[ly * Wc + x], fy00 = lfy[ly * Wc + x];
    if (x < Wc - 1)
      sdx += charb(fx00 - lfx[ly * Wc + x + 1]) + charb(fy00 - lfy[ly * Wc + x + 1]);
    if (y < Hc - 1)
      sdy += charb(fx00 - lfx[(ly + 1) * Wc + x]) + charb(fy00 - lfy[(ly + 1) * Wc + x]);
    if (y < Hc - 1 && x < Wc - 1) {
      sdr += charb(fx00 - lfx[(ly + 1) * Wc + x + 1]) +
             charb(fy00 - lfy[(ly + 1) * Wc + x + 1]);
      sur += charb(lfx[(ly + 1) * Wc + x] - lfx[ly * Wc + x + 1]) +
             charb(lfy[(ly + 1) * Wc + x] - lfy[ly * Wc + x + 1]);
    }
  }
  sdx = wave_sum32(sdx);
  sdy = wave_sum32(sdy);
  sdr = wave_sum32(sdr);
  sur = wave_sum32(sur);
  if ((threadIdx.x & 31) == 0) {
    float* a = &acc[ACC_SPAT + bp * 4];
    atomicAdd(a + 0, sdx);
    atomicAdd(a + 1, sdy);
    atomicAdd(a + 2, sdr);
    atomicAdd(a + 3, sur);
  }
}

// Temporal smoothing partials: warp next flow by current flow, L1-Charbonnier.
__global__ void k_temporal(const float* __restrict__ flow, float* __restrict__ acc) {
  const int bp = blockIdx.y;  // b*(Pc-1) + p
  const int b = bp / (Pc - 1);
  const int p = bp - b * (Pc - 1);
  const float* cfx = flow + (((size_t)(b * Pc + p)) * 2 + 0) * HWc;
  const float* cfy = cfx + HWc;
  const float* nfx = flow + (((size_t)(b * Pc + p + 1)) * 2 + 0) * HWc;
  const float* nfy = nfx + HWc;
  float sdt = 0.0f, smask = 0.0f;
  for (int i = blockIdx.x * blockDim.x + threadIdx.x; i < HWc;
       i += gridDim.x * blockDim.x) {
    int y = i / Wc, x = i - y * Wc;
    __builtin_prefetch(cfx + i + 4096, 0, 1);
    float cy = cfy[i], cx = cfx[i];
    float wy = (float)y + cy;
    float wx = (float)x + cx;
    float mask = (wy >= 0.0f && wy <= (float)(Hc - 1) &&
                  wx >= 0.0f && wx <= (float)(Wc - 1)) ? 1.0f : 0.0f;
    float y0f = floorf(wy), x0f = floorf(wx);
    float wy1 = wy - y0f, wy0 = 1.0f - wy1;
    float wx1 = wx - x0f, wx0 = 1.0f - wx1;
    int y0 = min(max((int)y0f, 0), Hc - 1);
    int y1 = min(max((int)y0f + 1, 0), Hc - 1);
    int x0 = min(max((int)x0f, 0), Wc - 1);
    int x1 = min(max((int)x0f + 1, 0), Wc - 1);
    float w00 = wy0 * wx0, w01 = wy0 * wx1, w10 = wy1 * wx0, w11 = wy1 * wx1;
    int i00 = y0 * Wc + x0, i01 = y0 * Wc + x1, i10 = y1 * Wc + x0, i11 = y1 * Wc + x1;
    float wfy = w00 * nfy[i00] + w01 * nfy[i01] + w10 * nfy[i10] + w11 * nfy[i11];
    float wfx = w00 * nfx[i00] + w01 * nfx[i01] + w10 * nfx[i10] + w11 * nfx[i11];
    float dt = sqrtf((cy - wfy) * (cy - wfy) + 1e-9f) +
               sqrtf((cx - wfx) * (cx - wfx) + 1e-9f);
    sdt += dt * mask;
    smask += mask;
  }
  sdt = wave_sum32(sdt);
  smask = wave_sum32(smask);
  if ((threadIdx.x & 31) == 0) {
    atomicAdd(&acc[ACC_TEMP + bp * 2 + 0], sdt);
    atomicAdd(&acc[ACC_TEMP + bp * 2 + 1], smask);
  }
}

// Combine partials into the scalar loss.
__global__ void k_final(const float* __restrict__ acc, float* __restrict__ out) {
  if (threadIdx.x != 0 || blockIdx.x != 0) return;
  float loss = 0.0f;
  // focus
  for (int b = 0; b < Bc; ++b)
    loss += acc[ACC_FOCUS_SQ + b] / (acc[ACC_FOCUS_CNT + b] + 1e-9f);
  // spatial: per-(b,p) means over per-term element counts, mean over p, /4
  const float cdx = (float)(Hc * (Wc - 1));
  const float cdy = (float)((Hc - 1) * Wc);
  const float cdg = (float)((Hc - 1) * (Wc - 1));
  float spat = 0.0f;
  for (int b = 0; b < Bc; ++b) {
    float sb = 0.0f;
    for (int p = 0; p < Pc; ++p) {
      const float* a = &acc[ACC_SPAT + (b * Pc + p) * 4];
      sb += a[0] / cdx + a[1] / cdy + a[2] / cdg + a[3] / cdg;
    }
    spat += sb / (4.0f * (float)Pc);
  }
  loss += 1.0f * spat;  // FLOW_SPAT_W
  // temporal
  float temp = 0.0f;
  for (int b = 0; b < Bc; ++b) {
    float tb = 0.0f;
    for (int p = 0; p < Pc - 1; ++p) {
      const float* a = &acc[ACC_TEMP + (b * (Pc - 1) + p) * 2];
      tb += a[0] / (a[1] + 1e-9f);
    }
    temp += tb / (float)(Pc - 1);
  }
  loss += 1.0f * temp;  // FLOW_TEMP_W
  out[0] = loss;
}

// ---------------- launcher ------------------------------------------
extern "C" void kernel_launch(void* const* d_in, const int* in_sizes, int n_in,
                              void* d_out, int out_size, void* d_ws, size_t ws_size,
                              hipStream_t stream) {
  (void)in_sizes; (void)n_in; (void)out_size; (void)ws_size;
  const float* ev   = (const float*)d_in[0];  // (B, N, 2)
  const float* pol  = (const float*)d_in[1];  // (B, 4N, 2)
  const float* ts   = (const float*)d_in[2];  // (B, 4N, 1)
  const float* flow = (const float*)d_in[3];  // (B, P, 2, H, W)

  float* ws  = (float*)d_ws;
  float* iwe = ws;                         // 4 planes * Bc * HWc
  float* acc = ws + 4 * (size_t)BHWc;      // ACC_SIZE floats

  const int nZero = 4 * BHWc + ACC_SIZE;
  k_zero<<<1024, 256, 0, stream>>>(ws, nZero);
  k_scatter<<<(Bc * Nc + 255) / 256, 256, 0, stream>>>(ev, pol, ts, iwe);
  k_focus<<<dim3(48, Bc), 256, 0, stream>>>(iwe, acc);
  k_spatial<<<dim3(Hc / TILE_H, Bc * Pc), 256, 0, stream>>>(flow, acc);
  k_temporal<<<dim3(48, Bc * (Pc - 1)), 256, 0, stream>>>(flow, acc);
  k_final<<<1, 32, 0, stream>>>(acc, (float*)d_out);
}